// Custom_Softmax_13228499272228
// MI455X (gfx1250) — compile-verified
//
#include <hip/hip_runtime.h>
#include <hip/hip_bf16.h>
#include <cstdint>

#define HEADS 16
#define SEQ   2048
#define ROWS_PER_BLOCK 8
#define WAVE  32

typedef __attribute__((ext_vector_type(2))) float v2f;
typedef __attribute__((ext_vector_type(4))) float v4f;
typedef __attribute__((ext_vector_type(8))) float v8f;
typedef __attribute__((ext_vector_type(4))) int   v4i;

union I4 { v4i v; int   i[4]; };
union F4 { v4f v; float f[4]; };

// One wave32 per (head,row). Stage the causal prefix of the row into LDS with
// CDNA5 async-to-LDS copies (non-temporal: single-use stream), two passes
// (max, exp+sum) over LDS, overwrite the staging with exp values, then stream
// requantized floats to HBM with non-temporal stores.
__global__ __launch_bounds__(256) void softmax_i8_rows(
    const int*   __restrict__ xq,        // (H*S*S) int32 (widened int8)
    const float* __restrict__ scale_x,   // (H*S)
    const float* __restrict__ scale_out, // (H*S)
    float*       __restrict__ out,       // (H*S*S) float (widened int8)
    float*       __restrict__ out_sc)    // (H*S) float  (pass-through so)
{
    __shared__ int smem[ROWS_PER_BLOCK * SEQ];   // 64 KB / block

    const int lane = threadIdx.x & (WAVE - 1);
    const int w    = threadIdx.x >> 5;
    const int g    = blockIdx.x * ROWS_PER_BLOCK + w;   // global row id in [0, H*S)
    const int row  = g & (SEQ - 1);                     // query index i (causal bound)

    const int  nvec = (row + 4) >> 2;                   // 16B chunks covering j<=row
    const int* rowp = xq + (size_t)g * SEQ;
    int*       srow = smem + w * SEQ;

    // scalar row scales: issue before async wait so latency overlaps
    const float sx = scale_x[g];
    const float so = scale_out[g];

    // ---- async global -> LDS staging (per-wave; no barrier needed) ----
    for (int c = lane; c < nvec; c += WAVE) {
        uint32_t lds = (uint32_t)(uintptr_t)(srow + c * 4);   // low 32b of flat LDS addr = LDS offset
        uint64_t ga  = (uint64_t)(uintptr_t)(rowp + c * 4);
        asm volatile("global_load_async_to_lds_b128 %0, %1, off th:TH_LOAD_NT"
                     :: "v"(lds), "v"(ga) : "memory");
    }
    asm volatile("s_wait_asynccnt 0" ::: "memory");

    // ---- pass 1: masked row max ----
    float m = -__builtin_inff();
    for (int c = lane; c < nvec; c += WAVE) {
        I4 q; q.v = *(const v4i*)(void*)(srow + c * 4);
        const int j0 = c * 4;
#pragma unroll
        for (int e = 0; e < 4; ++e) {
            float x = (float)q.i[e] * sx;
            if (j0 + e <= row) m = fmaxf(m, x);
        }
    }
#pragma unroll
    for (int off = 16; off >= 1; off >>= 1)
        m = fmaxf(m, __shfl_xor(m, off, WAVE));

    // ---- pass 2: exp + per-lane partial sum; overwrite LDS with exp ----
    float psum = 0.f;
    for (int c = lane; c < nvec; c += WAVE) {
        I4 q; q.v = *(const v4i*)(void*)(srow + c * 4);
        F4 ev;
        const int j0 = c * 4;
#pragma unroll
        for (int e = 0; e < 4; ++e) {
            float x  = (float)q.i[e] * sx;
            float ex = (j0 + e <= row) ? __expf(x - m) : 0.f;
            ev.f[e]  = ex;
            psum    += ex;
        }
        *(v4f*)(void*)(srow + c * 4) = ev.v;
    }

    // ---- cross-lane sum on the matrix pipe: D = ones(16x4) x B(4x16) ----
    // B VGPR0 holds psum per lane => every lane of D gets psum(n)+psum(n+16);
    // finish with 4 xor-shuffle adds. Full f32 precision, co-executes with VALU.
    v2f a; a[0] = 1.0f; a[1] = 1.0f;
    v2f b; b[0] = psum; b[1] = 0.0f;
    v8f cacc = {};
    cacc = __builtin_amdgcn_wmma_f32_16x16x4_f32(
        false, a, false, b, (short)0, cacc, false, false);
    float s = cacc[0];
    s += __shfl_xor(s, 1, WAVE);
    s += __shfl_xor(s, 2, WAVE);
    s += __shfl_xor(s, 4, WAVE);
    s += __shfl_xor(s, 8, WAVE);

    // softmax normalize fused with requant scale; v_rcp_f32 (1 ulp) instead of
    // the full IEEE div expansion — the standard quant-kernel choice.
    const float rs = __builtin_amdgcn_rcpf(s * so);

    // ---- pass 3: requant + stream full row out (zeros above diagonal) ----
    // Output is write-once / never re-read: non-temporal stores keep it out of L2.
    float* orow = out + (size_t)g * SEQ;
    for (int c = lane; c < SEQ / 4; c += WAVE) {
        F4 ov;
        if (c < nvec) {
            F4 ev; ev.v = *(const v4f*)(void*)(srow + c * 4);
            const int j0 = c * 4;
#pragma unroll
            for (int e = 0; e < 4; ++e) {
                float vq = rintf(ev.f[e] * rs);            // round-half-even == jnp.round
                vq = fminf(fmaxf(vq, -128.f), 127.f);
                ov.f[e] = (j0 + e <= row) ? vq : 0.f;
            }
        } else {
            ov.f[0] = ov.f[1] = ov.f[2] = ov.f[3] = 0.f;
        }
        __builtin_nontemporal_store(ov.v, (v4f*)(orow + c * 4));
    }

    if (lane == 0) out_sc[g] = so;   // second tuple output: so = scale_out[:, :S]
}

extern "C" void kernel_launch(void* const* d_in, const int* in_sizes, int n_in,
                              void* d_out, int out_size, void* d_ws, size_t ws_size,
                              hipStream_t stream) {
    const int*   xq = (const int*)  d_in[0];
    const float* sx = (const float*)d_in[1];
    const float* so = (const float*)d_in[2];
    float* out    = (float*)d_out;
    float* out_sc = out + (size_t)HEADS * SEQ * SEQ;   // tuple outputs concatenated

    const int total_rows = HEADS * SEQ;                // 32768
    dim3 grid(total_rows / ROWS_PER_BLOCK);            // 4096 blocks
    dim3 block(256);                                   // 8 wave32 per block
    softmax_i8_rows<<<grid, block, 0, stream>>>(xq, sx, so, out, out_sc);
}